// FSRU_53154515255536
// MI455X (gfx1250) — compile-verified
//
#include <hip/hip_runtime.h>
#include <math.h>

typedef float v2f __attribute__((ext_vector_type(2)));
typedef float v8f __attribute__((ext_vector_type(8)));

constexpr int BNUM   = 32;
constexpr int TLEN_C = 4096;
constexpr int DM     = 128;
constexpr int NSEG   = 512;
constexpr int HOP_C  = 256;
constexpr int NFR    = 17;
constexpr int NB     = 257;       // rfft bins of 512
constexpr int SFULL  = NB * NFR;  // 4369
constexpr int NPAT   = 196;
constexpr int NFQ    = 99;        // rfft bins of 196
constexpr int KPAD   = 260;       // 257 padded to mult of 4
constexpr int MPAD   = 272;       // 257 padded to mult of 16
constexpr float PI_C = 3.1415926f;

// ---- workspace layout (float offsets) ----
constexpr size_t OFF_WIN    = 0;
constexpr size_t OFF_NORM   = 512;
constexpr size_t OFF_WST_RE = 768;
constexpr size_t OFF_WST_IM = OFF_WST_RE + (size_t)MPAD * NSEG;
constexpr size_t OFF_CRE    = OFF_WST_IM + (size_t)MPAD * NSEG;
constexpr size_t OFF_CIM    = OFF_CRE + (size_t)NSEG * KPAD;
constexpr size_t OFF_WIR    = OFF_CIM + (size_t)NSEG * KPAD;
constexpr size_t OFF_WII    = OFF_WIR + (size_t)NFQ * NPAT;
constexpr size_t OFF_IIR    = OFF_WII + (size_t)NFQ * NPAT;
constexpr size_t OFF_III    = OFF_IIR + (size_t)NPAT * NFQ;
constexpr size_t OFF_ZFR    = OFF_III + (size_t)NPAT * NFQ;
constexpr size_t OFF_ZFI    = OFF_ZFR + (size_t)BNUM * NFQ * DM;
constexpr size_t OFF_GATE   = OFF_ZFI + (size_t)BNUM * NFQ * DM;
constexpr size_t OFF_YIMG   = OFF_GATE + (size_t)BNUM * DM;
constexpr size_t OFF_ZRE    = OFF_YIMG + (size_t)BNUM * NPAT * DM;   // per-b chunk
constexpr size_t OFF_ZIM    = OFF_ZRE + (size_t)NB * NFR * DM;

constexpr size_t TEXT_ELEMS = (size_t)BNUM * TLEN_C * DM;  // 16777216

__device__ __forceinline__ v8f wmma4(v2f a, v2f b, v8f c) {
  return __builtin_amdgcn_wmma_f32_16x16x4_f32(false, a, false, b, (short)0, c,
                                               false, false);
}

__device__ __forceinline__ float hannw(int t) {
  return 0.5f - 0.5f * cospif((float)t * (1.0f / 256.0f));
}

__device__ __forceinline__ float gelu_exact(float x) {
  return 0.5f * x * (1.0f + erff(x * 0.70710678f));
}

// =====================  table construction  =====================
__global__ void k_tables(float* __restrict__ ws) {
  int g = blockIdx.x * blockDim.x + threadIdx.x;
  int stride = gridDim.x * blockDim.x;
  float* win = ws + OFF_WIN;
  float* nrm = ws + OFF_NORM;
  float* wr  = ws + OFF_WST_RE;
  float* wi  = ws + OFF_WST_IM;
  float* cre = ws + OFF_CRE;
  float* cim = ws + OFF_CIM;
  float* fir = ws + OFF_WIR;
  float* fii = ws + OFF_WII;
  float* iir = ws + OFF_IIR;
  float* iii = ws + OFF_III;

  for (int i = g; i < NSEG; i += stride) win[i] = hannw(i);
  for (int i = g; i < HOP_C; i += stride) {
    float a = hannw(i), b = hannw(i + HOP_C);
    nrm[i] = a * a + b * b;
  }
  // STFT analysis matrix (M=freq padded to 272, K=time 512), win/256 folded in
  for (int i = g; i < MPAD * NSEG; i += stride) {
    int k = i >> 9, t = i & 511;
    float re = 0.f, im = 0.f;
    if (k <= 256) {
      int m = (k * t) & 511;
      float fr = (float)m * (1.0f / 256.0f);
      float sc = hannw(t) * (1.0f / 256.0f);
      re = cospif(fr) * sc;
      im = -sinpif(fr) * sc;
    }
    wr[i] = re;
    wi[i] = im;
  }
  // ISTFT synthesis matrix (M=time 512, K=freq padded to 260); irfft(1/512)
  // * WSUM(256) * win folded => win*0.5; Hermitian alpha; imag of DC/Nyquist
  // dropped (numpy c2r semantics).
  for (int i = g; i < NSEG * KPAD; i += stride) {
    int t = i / KPAD, k = i - t * KPAD;
    float re = 0.f, im = 0.f;
    if (k <= 256) {
      int m = (k * t) & 511;
      float fr = (float)m * (1.0f / 256.0f);
      bool edge = (k == 0) || (k == 256);
      float alpha = edge ? 1.0f : 2.0f;
      float wf = hannw(t) * 0.5f;
      re = alpha * cospif(fr) * wf;
      im = edge ? 0.0f : -alpha * sinpif(fr) * wf;
    }
    cre[i] = re;
    cim[i] = im;
  }
  // image forward rfft (ortho, /14)
  for (int i = g; i < NFQ * NPAT; i += stride) {
    int k = i / NPAT, n = i - k * NPAT;
    int m = (k * n) % NPAT;
    float fr = (float)m * (1.0f / 98.0f);
    fir[i] = cospif(fr) * (1.0f / 14.0f);
    fii[i] = -sinpif(fr) * (1.0f / 14.0f);
  }
  // image inverse rfft (ortho, /14, Hermitian alpha, imag DC/Nyq dropped)
  for (int i = g; i < NPAT * NFQ; i += stride) {
    int n = i / NFQ, k = i - n * NFQ;
    int m = (k * n) % NPAT;
    float fr = (float)m * (1.0f / 98.0f);
    bool edge = (k == 0) || (k == NFQ - 1);
    float alpha = edge ? 1.0f : 2.0f;
    iir[i] = alpha * cospif(fr) * (1.0f / 14.0f);
    iii[i] = edge ? 0.0f : -alpha * sinpif(fr) * (1.0f / 14.0f);
  }
}

// =====================  image branch  =====================
// forward DFT + filter bank -> Zf_img (B,99,D) complex
__global__ void k_img_fwd(const float* __restrict__ image,
                          const float* __restrict__ ifb,
                          float* __restrict__ ws) {
  int idx = blockIdx.x * 256 + threadIdx.x;
  if (idx >= BNUM * NFQ * DM) return;
  int d = idx & 127;
  int k = (idx >> 7) % NFQ;
  int b = idx / (NFQ * DM);
  const float* Wr = ws + OFF_WIR;
  const float* Wi = ws + OFF_WII;
  float sr = 0.f, si = 0.f;
  for (int n = 0; n < NPAT; n++) {
    float x = image[((size_t)b * NPAT + n) * DM + d];
    sr = fmaf(Wr[k * NPAT + n], x, sr);
    si = fmaf(Wi[k * NPAT + n], x, si);
  }
  const float c0 = cosf(PI_C), c1 = cosf(3.0f * PI_C);
  float fre = c0 * ifb[((size_t)(0 * NFQ + k) * DM + d) * 2 + 0] +
              c1 * ifb[((size_t)(1 * NFQ + k) * DM + d) * 2 + 0];
  float fim = c0 * ifb[((size_t)(0 * NFQ + k) * DM + d) * 2 + 1] +
              c1 * ifb[((size_t)(1 * NFQ + k) * DM + d) * 2 + 1];
  float zr = sr * sr - si * si;
  float zi = 2.0f * sr * si;
  const float inv = 1.0f / (float)NFQ;
  ws[OFF_ZFR + idx] = (zr * fre - zi * fim) * inv;
  ws[OFF_ZFI + idx] = (zr * fim + zi * fre) * inv;
}

// gate[b,e] = (mean_k Re(Zf*sel))[b,:] @ i2t_w^T + i2t_b
__global__ void k_gate(const float* __restrict__ sel,
                       const float* __restrict__ w,
                       const float* __restrict__ bias,
                       float* __restrict__ ws) {
  __shared__ float gb[DM];
  int b = blockIdx.x;
  int t = threadIdx.x;  // 128 threads
  float s = 0.f;
  for (int k = 0; k < NFQ; k++) {
    float zr = ws[OFF_ZFR + ((size_t)b * NFQ + k) * DM + t];
    float zi = ws[OFF_ZFI + ((size_t)b * NFQ + k) * DM + t];
    s += zr * sel[(k * DM + t) * 2 + 0] - zi * sel[(k * DM + t) * 2 + 1];
  }
  gb[t] = s * (1.0f / (float)NFQ);
  __syncthreads();
  float acc = bias[t];
  for (int d2 = 0; d2 < DM; d2++) acc += gb[d2] * w[t * DM + d2];
  ws[OFF_GATE + b * DM + t] = acc;
}

// inverse DFT of filtered image spectrum + residual -> y_img
__global__ void k_img_inv(const float* __restrict__ image,
                          float* __restrict__ ws) {
  int idx = blockIdx.x * 256 + threadIdx.x;
  if (idx >= BNUM * NPAT * DM) return;
  int d = idx & 127;
  int n = (idx >> 7) % NPAT;
  int b = idx / (NPAT * DM);
  const float* Ir = ws + OFF_IIR;
  const float* Ii = ws + OFF_III;
  float y = 0.f;
  for (int k = 0; k < NFQ; k++) {
    y = fmaf(Ir[n * NFQ + k], ws[OFF_ZFR + ((size_t)b * NFQ + k) * DM + d], y);
    y = fmaf(Ii[n * NFQ + k], ws[OFF_ZFI + ((size_t)b * NFQ + k) * DM + d], y);
  }
  y += image[idx];
  ws[OFF_YIMG + idx] = y;
}

// =====================  fused LN -> MLP(GELU) -> +res -> LN  =====================
// 256 threads; sx holds the 16x128 input rows (pre-LN1). Array references keep
// addrspace(3) visible so LDS ops stay ds_* (no generic null checks) and the
// weight streams stay global_load.
__device__ __forceinline__ void addnorm_block(
    float (&sx)[16 * DM], float (&sh)[16 * DM],
    const float* __restrict__ g1, const float* __restrict__ b1,
    const float* __restrict__ w1, const float* __restrict__ bb1,
    const float* __restrict__ w2, const float* __restrict__ bb2,
    const float* __restrict__ g2, const float* __restrict__ b2,
    float* __restrict__ outbase) {
  __shared__ float sm[16], sv[16];
  const int tid = threadIdx.x;
  const int lane = tid & 31, wave = tid >> 5;
  const int lr = lane & 15, hi = lane >> 4;
  const int n16 = wave * 16;
  __syncthreads();
  if (tid < 16) {
    float m = 0.f;
    for (int c = 0; c < DM; c++) m += sx[tid * DM + c];
    m *= (1.0f / 128.0f);
    float v = 0.f;
    for (int c = 0; c < DM; c++) {
      float dd = sx[tid * DM + c] - m;
      v += dd * dd;
    }
    sm[tid] = m;
    sv[tid] = v * (1.0f / 128.0f);
  }
  __syncthreads();
#pragma unroll
  for (int j = 0; j < 8; j++) {
    int e = j * 256 + tid;
    int r = e >> 7, c = e & 127;
    float x = (sx[e] - sm[r]) * rsqrtf(sv[r] + 1e-5f) * g1[c] + b1[c];
    sx[e] = x;
  }
  __syncthreads();
  // h1 = gelu(x @ w1 + bb1)
  v8f acc = {};
  for (int kk = 0; kk < 32; kk++) {
    int k0 = kk * 4 + hi * 2;
    v2f a = *(const v2f*)(&sx[lr * DM + k0]);
    v2f bv;
    bv[0] = w1[(size_t)k0 * DM + n16 + lr];
    bv[1] = w1[(size_t)(k0 + 1) * DM + n16 + lr];
    acc = wmma4(a, bv, acc);
  }
#pragma unroll
  for (int r = 0; r < 8; r++) {
    int ml = r + hi * 8;
    int c = n16 + lr;
    sh[ml * DM + c] = gelu_exact(acc[r] + bb1[c]);
  }
  __syncthreads();
  // h = h1 @ w2 + bb2 + x
  v8f acc2 = {};
  for (int kk = 0; kk < 32; kk++) {
    int k0 = kk * 4 + hi * 2;
    v2f a = *(const v2f*)(&sh[lr * DM + k0]);
    v2f bv;
    bv[0] = w2[(size_t)k0 * DM + n16 + lr];
    bv[1] = w2[(size_t)(k0 + 1) * DM + n16 + lr];
    acc2 = wmma4(a, bv, acc2);
  }
  float hv[8];
#pragma unroll
  for (int r = 0; r < 8; r++) {
    int ml = r + hi * 8;
    int c = n16 + lr;
    hv[r] = acc2[r] + bb2[c] + sx[ml * DM + c];
  }
  __syncthreads();
#pragma unroll
  for (int r = 0; r < 8; r++) {
    int ml = r + hi * 8;
    sh[ml * DM + n16 + lr] = hv[r];
  }
  __syncthreads();
  if (tid < 16) {
    float m = 0.f;
    for (int c = 0; c < DM; c++) m += sh[tid * DM + c];
    m *= (1.0f / 128.0f);
    float v = 0.f;
    for (int c = 0; c < DM; c++) {
      float dd = sh[tid * DM + c] - m;
      v += dd * dd;
    }
    sm[tid] = m;
    sv[tid] = v * (1.0f / 128.0f);
  }
  __syncthreads();
#pragma unroll
  for (int j = 0; j < 8; j++) {
    int e = j * 256 + tid;
    int r = e >> 7, c = e & 127;
    outbase[e] = (sh[e] - sm[r]) * rsqrtf(sv[r] + 1e-5f) * g2[c] + b2[c];
  }
}

// standalone addnorm (image branch), 16 rows per block
__global__ void k_addnorm(const float* __restrict__ y,
                          const float* __restrict__ g1,
                          const float* __restrict__ b1,
                          const float* __restrict__ w1,
                          const float* __restrict__ bb1,
                          const float* __restrict__ w2,
                          const float* __restrict__ bb2,
                          const float* __restrict__ g2,
                          const float* __restrict__ b2,
                          float* __restrict__ out) {
  __shared__ __align__(16) float sx[16 * DM];
  __shared__ __align__(16) float sh[16 * DM];
  size_t row0 = (size_t)blockIdx.x * 16;
#pragma unroll
  for (int j = 0; j < 8; j++) {
    int e = j * 256 + threadIdx.x;
    sx[e] = y[row0 * DM + e];
  }
  addnorm_block(sx, sh, g1, b1, w1, bb1, w2, bb2, g2, b2, out + row0 * DM);
}

// =====================  STFT (WMMA GEMM) + filter + gate, per batch b  =====================
__global__ void k_stft(const float* __restrict__ ecg,
                       const float* __restrict__ tfb, float* __restrict__ ws,
                       int b) {
  const int tid = threadIdx.x, lane = tid & 31, wave = tid >> 5;
  const int lr = lane & 15, hi = lane >> 4;
  const int f = blockIdx.x / 17;
  const int m16 = (blockIdx.x % 17) * 16;
  const int n16 = wave * 16;
  const int dcol = n16 + lr;
  const float* Wre = ws + OFF_WST_RE;
  const float* Wim = ws + OFF_WST_IM;
  const float* ecgb = ecg + (size_t)b * TLEN_C * DM;
  const int krow = m16 + lr;
  const int gbase = f * HOP_C - HOP_C;
  v8f ar = {}, ai = {};
  if (f >= 1 && f <= 15) {
    // interior frames: segment fully inside [0, TLEN) -> unguarded loads
    for (int kk = 0; kk < 128; kk++) {
      int t0 = kk * 4 + hi * 2;
      v2f a_re = *(const v2f*)(Wre + (size_t)krow * NSEG + t0);
      v2f a_im = *(const v2f*)(Wim + (size_t)krow * NSEG + t0);
      v2f bv;
      bv[0] = ecgb[(size_t)(gbase + t0) * DM + dcol];
      bv[1] = ecgb[(size_t)(gbase + t0 + 1) * DM + dcol];
      ar = wmma4(a_re, bv, ar);
      ai = wmma4(a_im, bv, ai);
    }
  } else {
    // frames 0 and 16: half the segment is zero padding
    for (int kk = 0; kk < 128; kk++) {
      int t0 = kk * 4 + hi * 2;
      v2f a_re = *(const v2f*)(Wre + (size_t)krow * NSEG + t0);
      v2f a_im = *(const v2f*)(Wim + (size_t)krow * NSEG + t0);
      int ga = gbase + t0, gb2 = ga + 1;
      v2f bv;
      bv[0] = (ga >= 0 && ga < TLEN_C) ? ecgb[(size_t)ga * DM + dcol] : 0.0f;
      bv[1] = (gb2 >= 0 && gb2 < TLEN_C) ? ecgb[(size_t)gb2 * DM + dcol] : 0.0f;
      ar = wmma4(a_re, bv, ar);
      ai = wmma4(a_im, bv, ai);
    }
  }
  const float c0 = cosf(PI_C), c1 = cosf(3.0f * PI_C);
  const float invS = 1.0f / (float)SFULL;
  const float gt = ws[OFF_GATE + b * DM + dcol];
  float* Zre = ws + OFF_ZRE;
  float* Zim = ws + OFF_ZIM;
#pragma unroll
  for (int r = 0; r < 8; r++) {
    int k = m16 + r + hi * 8;
    if (k <= 256) {
      int s = k * NFR + f;
      size_t t0i = ((size_t)(0 * SFULL + s) * DM + dcol) * 2;
      size_t t1i = ((size_t)(1 * SFULL + s) * DM + dcol) * 2;
      float fre = c0 * tfb[t0i + 0] + c1 * tfb[t1i + 0];
      float fim = c0 * tfb[t0i + 1] + c1 * tfb[t1i + 1];
      float zr = ar[r], zi = ai[r];
      float z2r = zr * zr - zi * zi;
      float z2i = 2.0f * zr * zi;
      float sc = invS * gt;
      size_t zo = ((size_t)k * NFR + f) * DM + dcol;
      Zre[zo] = (z2r * fre - z2i * fim) * sc;
      Zim[zo] = (z2r * fim + z2i * fre) * sc;
    }
  }
}

// ==========  fused ISTFT(WMMA) + OLA/norm + residual + addnorm, per batch b  ==========
__global__ void k_text(const float* __restrict__ ecg,
                       const float* __restrict__ g1,
                       const float* __restrict__ b1,
                       const float* __restrict__ w1,
                       const float* __restrict__ bb1,
                       const float* __restrict__ w2,
                       const float* __restrict__ bb2,
                       const float* __restrict__ g2,
                       const float* __restrict__ b2, float* __restrict__ out,
                       float* __restrict__ ws, int b) {
  __shared__ __align__(16) float sx[16 * DM];
  __shared__ __align__(16) float sh[16 * DM];
  const int tid = threadIdx.x, lane = tid & 31, wave = tid >> 5;
  const int lr = lane & 15, hi = lane >> 4;
  const int tau0 = blockIdx.x * 16;
  const int n16 = wave * 16;
  const int dcol = n16 + lr;
  const int f1 = tau0 >> 8;  // 0..15
  const int f2 = f1 + 1;     // 1..16
  const int bt2 = tau0 & 255;
  const float* Cre = ws + OFF_CRE;
  const float* Cim = ws + OFF_CIM;
  const float* Zre = ws + OFF_ZRE;
  const float* Zim = ws + OFF_ZIM;
  const int t1row = bt2 + 256 + lr;  // A rows for frame f1
  const int t2row = bt2 + lr;        // A rows for frame f2
  v8f acc = {};
  // K-steps 0..63 cover k = 0..255 -> always in range (unguarded)
  for (int kk = 0; kk < 64; kk++) {
    int k0 = kk * 4 + hi * 2;
    v2f aR1 = *(const v2f*)(Cre + (size_t)t1row * KPAD + k0);
    v2f aI1 = *(const v2f*)(Cim + (size_t)t1row * KPAD + k0);
    v2f aR2 = *(const v2f*)(Cre + (size_t)t2row * KPAD + k0);
    v2f aI2 = *(const v2f*)(Cim + (size_t)t2row * KPAD + k0);
    v2f bR1, bI1, bR2, bI2;
#pragma unroll
    for (int v = 0; v < 2; v++) {
      size_t o1 = ((size_t)(k0 + v) * NFR + f1) * DM + dcol;
      size_t o2 = ((size_t)(k0 + v) * NFR + f2) * DM + dcol;
      bR1[v] = Zre[o1];
      bI1[v] = Zim[o1];
      bR2[v] = Zre[o2];
      bI2[v] = Zim[o2];
    }
    acc = wmma4(aR1, bR1, acc);
    acc = wmma4(aI1, bI1, acc);
    acc = wmma4(aR2, bR2, acc);
    acc = wmma4(aI2, bI2, acc);
  }
  {  // tail K-step: k = 256..259, only k==256 exists (Nyquist)
    int k0 = 256 + hi * 2;
    v2f aR1 = *(const v2f*)(Cre + (size_t)t1row * KPAD + k0);
    v2f aI1 = *(const v2f*)(Cim + (size_t)t1row * KPAD + k0);
    v2f aR2 = *(const v2f*)(Cre + (size_t)t2row * KPAD + k0);
    v2f aI2 = *(const v2f*)(Cim + (size_t)t2row * KPAD + k0);
    v2f bR1 = {0.f, 0.f}, bI1 = {0.f, 0.f}, bR2 = {0.f, 0.f},
        bI2 = {0.f, 0.f};
    if (hi == 0) {
      size_t o1 = ((size_t)256 * NFR + f1) * DM + dcol;
      size_t o2 = ((size_t)256 * NFR + f2) * DM + dcol;
      bR1[0] = Zre[o1];
      bI1[0] = Zim[o1];
      bR2[0] = Zre[o2];
      bI2[0] = Zim[o2];
    }
    acc = wmma4(aR1, bR1, acc);
    acc = wmma4(aI1, bI1, acc);
    acc = wmma4(aR2, bR2, acc);
    acc = wmma4(aI2, bI2, acc);
  }
  const float* nrm = ws + OFF_NORM;
  const float* ecgb = ecg + (size_t)b * TLEN_C * DM;
#pragma unroll
  for (int r = 0; r < 8; r++) {
    int ml = r + hi * 8;
    int tau = tau0 + ml;
    float nv = nrm[tau & 255];
    nv = (nv > 1e-10f) ? nv : 1.0f;
    float y = acc[r] / nv + ecgb[(size_t)tau * DM + dcol];
    sx[ml * DM + dcol] = y;
  }
  addnorm_block(sx, sh, g1, b1, w1, bb1, w2, bb2, g2, b2,
                out + ((size_t)b * TLEN_C + tau0) * DM);
}

// =====================  launcher  =====================
extern "C" void kernel_launch(void* const* d_in, const int* in_sizes, int n_in,
                              void* d_out, int out_size, void* d_ws,
                              size_t ws_size, hipStream_t stream) {
  (void)in_sizes; (void)n_in; (void)out_size; (void)ws_size;
  const float* ecg     = (const float*)d_in[0];
  const float* image   = (const float*)d_in[1];
  const float* tfb     = (const float*)d_in[2];
  const float* ifb     = (const float*)d_in[3];
  const float* i2t_sel = (const float*)d_in[4];
  const float* i2t_w   = (const float*)d_in[5];
  const float* i2t_b   = (const float*)d_in[6];
  const float* t_ln1_g = (const float*)d_in[7];
  const float* t_ln1_b = (const float*)d_in[8];
  const float* t_w1    = (const float*)d_in[9];
  const float* t_b1    = (const float*)d_in[10];
  const float* t_w2    = (const float*)d_in[11];
  const float* t_b2    = (const float*)d_in[12];
  const float* t_ln2_g = (const float*)d_in[13];
  const float* t_ln2_b = (const float*)d_in[14];
  const float* i_ln1_g = (const float*)d_in[15];
  const float* i_ln1_b = (const float*)d_in[16];
  const float* i_w1    = (const float*)d_in[17];
  const float* i_b1    = (const float*)d_in[18];
  const float* i_w2    = (const float*)d_in[19];
  const float* i_b2    = (const float*)d_in[20];
  const float* i_ln2_g = (const float*)d_in[21];
  const float* i_ln2_b = (const float*)d_in[22];
  float* out = (float*)d_out;
  float* ws  = (float*)d_ws;

  k_tables<<<256, 256, 0, stream>>>(ws);

  // image branch
  k_img_fwd<<<(BNUM * NFQ * DM + 255) / 256, 256, 0, stream>>>(image, ifb, ws);
  k_gate<<<BNUM, 128, 0, stream>>>(i2t_sel, i2t_w, i2t_b, ws);
  k_img_inv<<<(BNUM * NPAT * DM + 255) / 256, 256, 0, stream>>>(image, ws);
  k_addnorm<<<(BNUM * NPAT) / 16, 256, 0, stream>>>(
      ws + OFF_YIMG, i_ln1_g, i_ln1_b, i_w1, i_b1, i_w2, i_b2, i_ln2_g,
      i_ln2_b, out + TEXT_ELEMS);

  // ecg branch, chunked per batch to keep spectral scratch tiny
  for (int b = 0; b < BNUM; b++) {
    k_stft<<<17 * 17, 256, 0, stream>>>(ecg, tfb, ws, b);
    k_text<<<TLEN_C / 16, 256, 0, stream>>>(ecg, t_ln1_g, t_ln1_b, t_w1, t_b1,
                                            t_w2, t_b2, t_ln2_g, t_ln2_b, out,
                                            ws, b);
  }
}